// PointDistanceLoss_45844480917997
// MI455X (gfx1250) — compile-verified
//
#include <hip/hip_runtime.h>

typedef __attribute__((ext_vector_type(2))) float v2f;
typedef __attribute__((ext_vector_type(8))) float v8f;

namespace {
constexpr int kB     = 4;
constexpr int kN     = 8192;
constexpr int kTiles = kN / 16;          // 512 tiles per axis
constexpr int kWaves = 8;                // 256-thread block = 8 x wave32
constexpr int kIters = kTiles / kWaves;  // 64 j-tiles per wave (compile-time)
}

__global__ __launch_bounds__(64)
void pdl_zero_kernel(float* __restrict__ out, int n) {
  int i = blockIdx.x * blockDim.x + threadIdx.x;
  if (i < n) out[i] = 0.0f;
}

// One block per (batch, i-tile). Each of the 8 waves owns a fixed 16-row A tile
// of pred and sweeps 64 of the 512 16-col B tiles of target.
//
// d2[m][n] = p2[m] + t2[n] - 2*p.t  via one V_WMMA_F32_16X16X4_F32:
//   A[m][:] = (-2px, -2py, -2pz, p2[m])   (16x4)
//   B[:][n] = ( tx ,  ty ,  tz , 1.0  )   (4x16)
// then per element: + t2[n], clamp at 0, sqrt, accumulate.
__global__ __launch_bounds__(256)
void pdl_wmma_kernel(const float* __restrict__ pred,
                     const float* __restrict__ targ,
                     float* __restrict__ out) {
  const int b     = blockIdx.x / kTiles;
  const int itile = blockIdx.x - b * kTiles;
  const int lane  = threadIdx.x & 31;
  const int wave  = threadIdx.x >> 5;     // 0..7
  const int lidx  = lane & 15;            // row/col index within tile
  const int lhalf = lane >> 4;            // 0: K={0,1}; 1: K={2,3}

  const float* __restrict__ pb = pred + (size_t)b * kN * 3;
  const float* __restrict__ tb = targ + (size_t)b * kN * 3;

  // Build the A-matrix fragment (16x4 f32, 2 VGPRs/lane).
  // Lane L: M = L%16; VGPR j holds K = 2*(L>>4) + j.
  v2f amat;
  {
    const int m  = itile * 16 + lidx;
    const float px = pb[3 * m + 0];
    const float py = pb[3 * m + 1];
    const float pz = pb[3 * m + 2];
    const float p2 = px * px + py * py + pz * pz;
    amat.x = lhalf ? (-2.0f * pz) : (-2.0f * px);   // K=2 : K=0
    amat.y = lhalf ? p2           : (-2.0f * py);   // K=3 : K=1
  }

  v8f acc = {};   // per-lane running sum of sqrt distances (8 D rows)

  // Compile-time trip count + pointer bump -> scalar loop counter, EXEC
  // stays all-1s (required for WMMA), no per-iteration vector compare.
  const float* __restrict__ tp = tb + 3 * (wave * 16 + lidx);
  constexpr int kStride = 3 * 16 * kWaves;   // floats between j-tiles of a wave

#pragma unroll 2
  for (int it = 0; it < kIters; ++it) {
    const float tx = tp[0];
    const float ty = tp[1];
    const float tz = tp[2];
    tp += kStride;
    const float t2 = tx * tx + ty * ty + tz * tz;

    // B-matrix fragment (4x16 f32): lane L: N = L%16; VGPR j -> K = 2*(L>>4)+j.
    v2f bmat;
    bmat.x = lhalf ? tz   : tx;    // K=2 : K=0
    bmat.y = lhalf ? 1.0f : ty;    // K=3 : K=1

    v8f c = {};
    // (neg_a, A, neg_b, B, c_mod, C, reuse_a, reuse_b)
    c = __builtin_amdgcn_wmma_f32_16x16x4_f32(
        false, amat, false, bmat, (short)0, c, false, false);

    // D VGPR r: column N = lidx for every lane -> t2 is the right per-lane add.
#pragma unroll
    for (int r = 0; r < 8; ++r) {
      float d2 = c[r] + t2;
      d2 = fmaxf(d2, 0.0f);
      acc[r] += __builtin_amdgcn_sqrtf(d2);
    }
  }

  // Lane-local fold of the 8 accumulator rows.
  float s = (acc[0] + acc[1]) + (acc[2] + acc[3]) +
            (acc[4] + acc[5]) + (acc[6] + acc[7]);

  // wave32 butterfly reduction.
#pragma unroll
  for (int off = 16; off > 0; off >>= 1) {
    s += __shfl_xor(s, off, 32);
  }

  // Cross-wave reduction in LDS, one atomic per block.
  __shared__ float wsum[kWaves];
  if (lane == 0) wsum[wave] = s;
  __syncthreads();
  if (threadIdx.x == 0) {
    float tot = 0.0f;
#pragma unroll
    for (int w = 0; w < kWaves; ++w) tot += wsum[w];
    const float scale = 1.0f / ((float)kN * (float)kN);
    atomicAdd(&out[b], tot * scale);
  }
}

extern "C" void kernel_launch(void* const* d_in, const int* in_sizes, int n_in,
                              void* d_out, int out_size, void* d_ws, size_t ws_size,
                              hipStream_t stream) {
  (void)in_sizes; (void)n_in; (void)d_ws; (void)ws_size;
  const float* pred = (const float*)d_in[0];
  const float* targ = (const float*)d_in[1];
  float* out = (float*)d_out;

  // d_out is poisoned by the harness; zero it before accumulating.
  pdl_zero_kernel<<<1, 64, 0, stream>>>(out, out_size);

  const int grid = kB * kTiles;   // 2048 blocks
  pdl_wmma_kernel<<<grid, 256, 0, stream>>>(pred, targ, out);
}